// GNNBlock_7052336300588
// MI455X (gfx1250) — compile-verified
//
#include <hip/hip_runtime.h>

typedef __attribute__((ext_vector_type(2))) float v2f;
typedef __attribute__((ext_vector_type(8))) float v8f;

#define BG    512   // graphs (blocks)
#define NN    128   // nodes per graph
#define FIN   128
#define CH    256   // HEADS*HID
#define HIDC  64
#define SA    258   // bufA row stride (padded, even -> b64-aligned, conflict-free)
#define SBT   130   // bufBT row stride (padded)
#define SW    34    // W-stage row stride (padded)

__device__ __forceinline__ float lrelu02(float v) { return v >= 0.0f ? v : 0.2f * v; }

__global__ __launch_bounds__(512, 1)
void gat3_kernel(const float* __restrict__ x,
                 const float* __restrict__ W0, const float* __restrict__ as0,
                 const float* __restrict__ ad0, const float* __restrict__ bias0,
                 const float* __restrict__ W1, const float* __restrict__ as1,
                 const float* __restrict__ ad1, const float* __restrict__ bias1,
                 const float* __restrict__ W2, const float* __restrict__ as2,
                 const float* __restrict__ ad2, const float* __restrict__ bias2,
                 float* __restrict__ out)
{
    __shared__ float bufA[NN * SA];    // layer input, node-major          (129 KB)
    __shared__ float bufBT[CH * SBT];  // h, stored transposed [col][node] (130 KB)
    __shared__ float wT[CH * SW];      // W chunk transposed [n][k]        ( 34 KB)
    __shared__ float aSrc[NN * 4];
    __shared__ float aDst[NN * 4];
    __shared__ float rMax[NN * 4];
    __shared__ float rInv[NN * 4];
    __shared__ float sMax[8];
    __shared__ float attS[CH];
    __shared__ float attD[CH];

    const int t     = threadIdx.x;
    const int b     = blockIdx.x;
    const int wave  = t >> 5;
    const int lane  = t & 31;
    const int lhalf = lane >> 4;   // lane half: K/M split of WMMA fragments
    const int l16   = lane & 15;

    // ---- stage x -> bufA (coalesced global reads) ----
    for (int idx = t; idx < NN * FIN; idx += 512) {
        const int r = idx >> 7, c = idx & 127;
        bufA[r * SA + c] = x[(b * NN + r) * FIN + c];
    }
    __syncthreads();

    const float* Wl[3]  = {W0, W1, W2};
    const float* ASl[3] = {as0, as1, as2};
    const float* ADl[3] = {ad0, ad1, ad2};
    const float* Bl[3]  = {bias0, bias1, bias2};
    const int    Kl[3]  = {FIN, CH, CH};

    for (int layer = 0; layer < 3; ++layer) {
        const int K = Kl[layer];
        const float* __restrict__ W = Wl[layer];

        // ================= GEMM1: h[128,256] = A[128,K] @ W[K,256] =================
        // wave -> M-tile (wave&7), 8 N-tiles starting at (wave>>3)*8
        {
            const int mt = wave & 7;
            const int nb = (wave >> 3) * 8;
            v8f acc[8];
            #pragma unroll
            for (int nt = 0; nt < 8; ++nt)
                #pragma unroll
                for (int e = 0; e < 8; ++e) acc[nt][e] = 0.0f;

            const float* aRow = bufA + (mt * 16 + l16) * SA;

            for (int k0 = 0; k0 < K; k0 += 32) {
                // stage W[k0..k0+31][0..255] transposed into wT (coalesced global)
                for (int idx = t; idx < 32 * CH; idx += 512) {
                    const int kk = idx >> 8, n = idx & 255;
                    wT[n * SW + kk] = W[(k0 + kk) * CH + n];
                }
                __syncthreads();
                #pragma unroll
                for (int ks = 0; ks < 32; ks += 4) {
                    // A frag: lane(m=l16): V0/V1 = A[m][k0+ks+2*lhalf +0/+1] (contiguous)
                    v2f af = *(const v2f*)(aRow + k0 + ks + 2 * lhalf);
                    #pragma unroll
                    for (int nt = 0; nt < 8; ++nt) {
                        const int ncol = (nb + nt) * 16 + l16;
                        // B frag: V0/V1 = W[k][n], W[k+1][n] -> contiguous in wT
                        v2f bf = *(const v2f*)(wT + ncol * SW + ks + 2 * lhalf);
                        acc[nt] = __builtin_amdgcn_wmma_f32_16x16x4_f32(
                            false, af, false, bf, (short)0, acc[nt], false, false);
                    }
                }
                __syncthreads();
            }

            // store h transposed: bufBT[col*SBT + row]; D vgpr e -> row mt*16+8*lhalf+e
            #pragma unroll
            for (int nt = 0; nt < 8; ++nt) {
                const int col = (nb + nt) * 16 + l16;
                float* dst = bufBT + col * SBT + mt * 16 + 8 * lhalf;
                #pragma unroll
                for (int e = 0; e < 8; e += 2) {
                    v2f p = {acc[nt][e], acc[nt][e + 1]};
                    *(v2f*)(dst + e) = p;
                }
            }
        }
        __syncthreads();

        // ================= attention scalars + softmax stats =================
        if (t < CH) { attS[t] = ASl[layer][t]; attD[t] = ADl[layer][t]; }
        __syncthreads();
        {   // a_src/a_dst: thread -> (head = t>>7, node i = t&127)
            const int hd = t >> 7, i = t & 127;
            float ss = 0.0f, dd = 0.0f;
            for (int c = 0; c < HIDC; ++c) {
                const float hv = bufBT[(hd * HIDC + c) * SBT + i];
                ss += hv * attS[hd * HIDC + c];
                dd += hv * attD[hd * HIDC + c];
            }
            aSrc[i * 4 + hd] = ss;
            aDst[i * 4 + hd] = dd;
        }
        __syncthreads();
        if (t < 4) {   // max_j a_src per head (leaky is monotone -> rowmax = leaky(smax+adst))
            float m = -1e30f;
            for (int j = 0; j < NN; ++j) m = fmaxf(m, aSrc[j * 4 + t]);
            sMax[t] = m;
        }
        __syncthreads();
        {   // per (i,head): row max + 1/denominator
            const int hd = t >> 7, i = t & 127;
            const float ad = aDst[i * 4 + hd];
            const float rm = lrelu02(sMax[hd] + ad);
            float s = 0.0f;
            for (int j = 0; j < NN; ++j)
                s += __expf(lrelu02(aSrc[j * 4 + hd] + ad) - rm);
            rMax[i * 4 + hd] = rm;
            rInv[i * 4 + hd] = 1.0f / s;
        }
        __syncthreads();

        // ================= GEMM2: out[i,hc] = sum_j alpha[i,j,h] * h[j,hc] ========
        // wave -> M-tile (wave&7), head pair (wave>>3): 2 heads x 4 N-tiles
        {
            const int mt = wave & 7;
            const int h0 = (wave >> 3) * 2;
            const int mrow = mt * 16 + l16;
            const float* __restrict__ bias = Bl[layer];

            const float ad0v = aDst[mrow * 4 + h0];
            const float ad1v = aDst[mrow * 4 + h0 + 1];
            const float rm0  = rMax[mrow * 4 + h0];
            const float rm1  = rMax[mrow * 4 + h0 + 1];
            const float ri0  = rInv[mrow * 4 + h0];
            const float ri1  = rInv[mrow * 4 + h0 + 1];

            v8f acc[2][4];
            #pragma unroll
            for (int hh = 0; hh < 2; ++hh)
                #pragma unroll
                for (int nt = 0; nt < 4; ++nt)
                    #pragma unroll
                    for (int e = 0; e < 8; ++e) acc[hh][nt][e] = 0.0f;

            for (int ks = 0; ks < NN; ks += 4) {
                const int j0 = ks + 2 * lhalf;
                // a_src for j0,j0+1 and both heads (heads adjacent -> v2f loads)
                v2f s0 = *(const v2f*)(aSrc + j0 * 4 + h0);
                v2f s1 = *(const v2f*)(aSrc + (j0 + 1) * 4 + h0);
                v2f af0, af1;   // alpha fragments computed on the fly
                af0[0] = __expf(lrelu02(s0[0] + ad0v) - rm0) * ri0;
                af0[1] = __expf(lrelu02(s1[0] + ad0v) - rm0) * ri0;
                af1[0] = __expf(lrelu02(s0[1] + ad1v) - rm1) * ri1;
                af1[1] = __expf(lrelu02(s1[1] + ad1v) - rm1) * ri1;
                #pragma unroll
                for (int hh = 0; hh < 2; ++hh) {
                    const v2f af = hh ? af1 : af0;
                    const int cb = (h0 + hh) * HIDC;
                    #pragma unroll
                    for (int nt = 0; nt < 4; ++nt) {
                        const int col = cb + nt * 16 + l16;
                        v2f bf = *(const v2f*)(bufBT + col * SBT + j0); // h[j][col], h[j+1][col]
                        acc[hh][nt] = __builtin_amdgcn_wmma_f32_16x16x4_f32(
                            false, af, false, bf, (short)0, acc[hh][nt], false, false);
                    }
                }
            }

            // bias add + store next-layer input node-major into bufA
            #pragma unroll
            for (int hh = 0; hh < 2; ++hh) {
                const int cb = (h0 + hh) * HIDC;
                #pragma unroll
                for (int nt = 0; nt < 4; ++nt) {
                    const int col = cb + nt * 16 + l16;
                    const float bv = bias[col];
                    #pragma unroll
                    for (int e = 0; e < 8; ++e) {
                        const int row = mt * 16 + 8 * lhalf + e;
                        bufA[row * SA + col] = acc[hh][nt][e] + bv;
                    }
                }
            }
        }
        __syncthreads();
    } // layers

    // ---- final: out[b,f] = sum_i h[i,f] ----
    if (t < CH) {
        float s = 0.0f;
        for (int i = 0; i < NN; ++i) s += bufA[i * SA + t];
        out[b * CH + t] = s;
    }
}

extern "C" void kernel_launch(void* const* d_in, const int* in_sizes, int n_in,
                              void* d_out, int out_size, void* d_ws, size_t ws_size,
                              hipStream_t stream) {
    (void)in_sizes; (void)n_in; (void)d_ws; (void)ws_size; (void)out_size;
    const float* x   = (const float*)d_in[0];
    // d_in[1] = batch_mask (layout is uniform: graph b owns nodes [b*128,(b+1)*128)) - unused
    const float* W0  = (const float*)d_in[2];
    const float* as0 = (const float*)d_in[3];
    const float* ad0 = (const float*)d_in[4];
    const float* b0  = (const float*)d_in[5];
    const float* W1  = (const float*)d_in[6];
    const float* as1 = (const float*)d_in[7];
    const float* ad1 = (const float*)d_in[8];
    const float* b1  = (const float*)d_in[9];
    const float* W2  = (const float*)d_in[10];
    const float* as2 = (const float*)d_in[11];
    const float* ad2 = (const float*)d_in[12];
    const float* b2  = (const float*)d_in[13];
    float* out = (float*)d_out;

    gat3_kernel<<<dim3(BG), dim3(512), 0, stream>>>(
        x, W0, as0, ad0, b0, W1, as1, ad1, b1, W2, as2, ad2, b2, out);
}